// Encoder_64201171141317
// MI455X (gfx1250) — compile-verified
//
#include <hip/hip_runtime.h>
#include <hip/hip_bf16.h>
#include <math.h>

// ---------------------------------------------------------------------------
// Problem constants (match reference)
// ---------------------------------------------------------------------------
#define FEAT_DIM   128
#define KDIM       256          // 2 * FEAT_DIM
#define EMBED_DIM  128
#define BATCH      8192
#define NUM_SAMPLE 10
#define N_LABELED  16384
#define INV_TEMP   0.2f         // 1 / 5.0
#define EPSV       1e-6f

typedef __attribute__((ext_vector_type(2))) float v2f;
typedef __attribute__((ext_vector_type(8))) float v8f;

// ---------------------------------------------------------------------------
// Kernel 1: fused gather (self + neighbor-mean) -> LDS tile -> f32 WMMA GEMM
// out[m][n] = relu( sum_k W[m][k] * combined[n][k] )
// Block: 256 threads (8 wave32), handles one 16-wide batch strip (all 8 M tiles)
// Grid: BATCH/16 = 512 blocks
// ---------------------------------------------------------------------------
__global__ __launch_bounds__(256) void fused_gather_gemm(
    const int* __restrict__ nodes,
    const int* __restrict__ neigh,        // [BATCH][NUM_SAMPLE]
    const float* __restrict__ feat,       // [N_NODES][FEAT_DIM]
    const float* __restrict__ W,          // [EMBED_DIM][KDIM]
    float* __restrict__ out)              // [EMBED_DIM][BATCH]
{
    // padded stride 260: bank = (row*260 + k) % 64 -> 4*row spread, conflict free
    __shared__ float lds[16][260];

    const int nb  = blockIdx.x * 16;      // first batch row of this strip
    const int tid = threadIdx.x;

    // ---- stage combined[nb+r][0:256] into LDS ----
    {
        const int d = tid & 127;          // feature dim
        for (int r = tid >> 7; r < 16; r += 2) {
            const int n = nb + r;
            // self features -> cols [0,128)
            lds[r][d] = feat[(size_t)nodes[n] * FEAT_DIM + d];
            // neighbor mean -> cols [128,256)
            const int* ni = neigh + (size_t)n * NUM_SAMPLE;
            float acc = 0.f;
#pragma unroll
            for (int s = 0; s < NUM_SAMPLE; ++s)
                acc += feat[(size_t)ni[s] * FEAT_DIM + d];
            lds[r][FEAT_DIM + d] = acc * (1.0f / NUM_SAMPLE);
        }
    }
    __syncthreads();

    // ---- per-wave 16x16 WMMA tile, K = 256 in steps of 4 ----
    const int wave = tid >> 5;            // 0..7 -> M tile
    const int lane = tid & 31;
    const int row  = lane & 15;           // M index (A) / N index (B)
    const int h    = lane >> 4;           // lane half: K offset +2

    const int m0 = wave * 16;
    const float* wrow = W + (size_t)(m0 + row) * KDIM;

    v8f acc = {};
#pragma unroll 8
    for (int k0 = 0; k0 < KDIM; k0 += 4) {
        // A frag: VGPR0 = {K=k0+2h}, VGPR1 = {K=k0+2h+1}  (ISA 16x4 f32 layout)
        v2f a = *(const v2f*)(wrow + k0 + 2 * h);
        // B frag: same K striping, lane = column (batch row of strip)
        v2f b = *(const v2f*)(&lds[row][k0 + 2 * h]);
        acc = __builtin_amdgcn_wmma_f32_16x16x4_f32(
            /*neg_a=*/false, a, /*neg_b=*/false, b,
            /*c_mod=*/(short)0, acc, /*reuse_a=*/false, /*reuse_b=*/false);
    }

    // ---- store with ReLU; C/D layout: VGPR j, lane half h -> M = j + 8h ----
#pragma unroll
    for (int j = 0; j < 8; ++j) {
        const int m = m0 + j + 8 * h;
        out[(size_t)m * BATCH + nb + row] = fmaxf(acc[j], 0.f);
    }
}

// ---------------------------------------------------------------------------
// Kernel 2: zero the atomic accumulators (denom[2] + accum_pos[128] + accum_neg[128])
// ---------------------------------------------------------------------------
__global__ void vec_init(float* __restrict__ p)
{
    for (int i = threadIdx.x; i < 258; i += 256) p[i] = 0.f;
}

// ---------------------------------------------------------------------------
// Kernel 3: per-row logits e[m] = exp(cos(x_m, vec)/TEMP) + atomic denom
// Block: 128 threads = 4 wave32, one labeled row per wave. Grid: 16384/4.
// ---------------------------------------------------------------------------
__global__ __launch_bounds__(128) void vec_logits(
    const int* __restrict__ idx,
    const float* __restrict__ feat,
    const float* __restrict__ vec,        // [FEAT_DIM]
    float* __restrict__ e,                // [N_LABELED]
    float* __restrict__ denom)            // single float
{
    const int lane = threadIdx.x & 31;
    const int wv   = threadIdx.x >> 5;
    const int m    = blockIdx.x * 4 + wv;

    const float* x = feat + (size_t)idx[m] * FEAT_DIM;

    float dot = 0.f, sq = 0.f, vsq = 0.f;
#pragma unroll
    for (int i = 0; i < FEAT_DIM / 32; ++i) {
        const float xv = x[lane + 32 * i];
        const float vv = vec[lane + 32 * i];
        dot += xv * vv;
        sq  += xv * xv;
        vsq += vv * vv;
    }
#pragma unroll
    for (int off = 16; off > 0; off >>= 1) {
        dot += __shfl_xor(dot, off, 32);
        sq  += __shfl_xor(sq,  off, 32);
        vsq += __shfl_xor(vsq, off, 32);
    }
    if (lane == 0) {
        const float nv = fmaxf(sqrtf(vsq), EPSV);
        const float nx = fmaxf(sqrtf(sq),  EPSV);
        const float c  = dot / (nv * nx);          // cos in [-1,1] -> exp safe
        const float ee = __expf(c * INV_TEMP);
        e[m] = ee;
        atomicAdd(denom, ee);
    }
}

// ---------------------------------------------------------------------------
// Kernel 4: weighted feature sum: accum[d] += sum_m e[m] * x[m][d]
// Block: 128 threads (one per dim), each block handles 128 labeled rows.
// ---------------------------------------------------------------------------
__global__ __launch_bounds__(128) void vec_accum(
    const int* __restrict__ idx,
    const float* __restrict__ feat,
    const float* __restrict__ e,
    float* __restrict__ accum)            // [FEAT_DIM]
{
    const int d  = threadIdx.x;
    const int m0 = blockIdx.x * 128;
    float local = 0.f;
    for (int r = 0; r < 128; ++r) {
        const int m = m0 + r;
        local += e[m] * feat[(size_t)idx[m] * FEAT_DIM + d];
    }
    atomicAdd(&accum[d], local);
}

// ---------------------------------------------------------------------------
// Kernel 5: finalize new_pos / new_neg = accum / denom
// ---------------------------------------------------------------------------
__global__ void vec_final(
    const float* __restrict__ acc_p, const float* __restrict__ acc_n,
    const float* __restrict__ denom,
    float* __restrict__ outp, float* __restrict__ outn)
{
    const int t = threadIdx.x;
    if (t < FEAT_DIM)        outp[t] = acc_p[t] / denom[0];
    else                     outn[t - FEAT_DIM] = acc_n[t - FEAT_DIM] / denom[1];
}

// ---------------------------------------------------------------------------
// Launch
// ---------------------------------------------------------------------------
extern "C" void kernel_launch(void* const* d_in, const int* in_sizes, int n_in,
                              void* d_out, int out_size, void* d_ws, size_t ws_size,
                              hipStream_t stream)
{
    const int*   nodes = (const int*)  d_in[0];   // [8192]
    const int*   neigh = (const int*)  d_in[1];   // [8192,10]
    const int*   posi  = (const int*)  d_in[2];   // [16384]
    const int*   negi  = (const int*)  d_in[3];   // [16384]
    const float* feat  = (const float*)d_in[4];   // [1e6,128]
    const float* W     = (const float*)d_in[5];   // [128,256]
    const float* pvec  = (const float*)d_in[6];   // [1,128]
    const float* nvec  = (const float*)d_in[7];   // [1,128]

    float* out  = (float*)d_out;                       // [128,8192]
    float* outp = out + (size_t)EMBED_DIM * BATCH;     // [128]
    float* outn = outp + FEAT_DIM;                     // [128]

    // workspace layout (floats): e_pos[16384] | e_neg[16384] | denom[2] | acc_p[128] | acc_n[128]
    float* ws    = (float*)d_ws;
    float* e_pos = ws;
    float* e_neg = ws + N_LABELED;
    float* denom = ws + 2 * N_LABELED;
    float* acc_p = denom + 2;
    float* acc_n = acc_p + FEAT_DIM;

    // big fused gather + WMMA GEMM (independent of the vector path)
    fused_gather_gemm<<<BATCH / 16, 256, 0, stream>>>(nodes, neigh, feat, W, out);

    // softmax-weighted vector updates
    vec_init<<<1, 256, 0, stream>>>(denom);
    vec_logits<<<N_LABELED / 4, 128, 0, stream>>>(posi, feat, pvec, e_pos, denom + 0);
    vec_logits<<<N_LABELED / 4, 128, 0, stream>>>(negi, feat, nvec, e_neg, denom + 1);
    vec_accum<<<N_LABELED / 128, 128, 0, stream>>>(posi, feat, e_pos, acc_p);
    vec_accum<<<N_LABELED / 128, 128, 0, stream>>>(negi, feat, e_neg, acc_n);
    vec_final<<<1, 256, 0, stream>>>(acc_p, acc_n, denom, outp, outn);
}